// TFModel_8400956031318
// MI455X (gfx1250) — compile-verified
//
#include <hip/hip_runtime.h>

// Problem constants (from the reference)
#define PH_   7
#define PW_   7
#define SP_   4
#define C_    10
#define FH_   34
#define FW_   34
#define NPIX  (FH_ * FW_)     // 1156 pixels per (bin) plane
#define BINS  (PH_ * PW_)     // 49
#define CPAD  12              // pad 10 channels -> 12 floats (48B, 16B aligned)
#define TPB   256             // 8 waves (wave32)

typedef __attribute__((address_space(1))) int gas_int;   // global (AS1)
typedef __attribute__((address_space(3))) int las_int;   // LDS (AS3)

__global__ __launch_bounds__(TPB) void psroi_align_kernel(
    const float* __restrict__ ft,    // (C*49, 34, 34) float32
    const float* __restrict__ rois,  // (N, 5) float32
    float* __restrict__ out,         // (N, C, 49) float32
    int N)
{
  // Per-bin feature slab, channel-innermost, padded: slab[pix*12 + c]
  __shared__ __align__(16) float slab[NPIX * CPAD];

  const int p   = blockIdx.y;          // bin index 0..48
  const int ph  = p / PW_;
  const int pw  = p - ph * PW_;
  const int tid = threadIdx.x;

  // ---- Stage this bin's 45KB feature slab global -> LDS (transposing copy).
  // Global layout: ft[(c*49 + p)*1156 + pix]; LDS layout: slab[pix*12 + c].
  for (int i = tid; i < C_ * NPIX; i += TPB) {
    const int c   = i / NPIX;
    const int pix = i - c * NPIX;
    const float* gsrc = ft + (size_t)(c * BINS + p) * NPIX + pix;
    float*       ldst = &slab[pix * CPAD + c];
#if __has_builtin(__builtin_amdgcn_global_load_async_to_lds_b32)
    // CDNA5 async DMA path: per-lane global->LDS copy, tracked by ASYNCcnt.
    __builtin_amdgcn_global_load_async_to_lds_b32(
        (gas_int*)gsrc, (las_int*)ldst, /*offset=*/0, /*cpol=*/0);
#else
    *ldst = *gsrc;  // fallback: global_load + ds_store
#endif
  }
#if __has_builtin(__builtin_amdgcn_global_load_async_to_lds_b32)
#if __has_builtin(__builtin_amdgcn_s_wait_asynccnt)
  __builtin_amdgcn_s_wait_asynccnt(0);
#else
  asm volatile("s_wait_asynccnt 0x0" ::: "memory");
#endif
#endif
  __syncthreads();

  const int r = blockIdx.x * TPB + tid;   // one ROI per thread
  if (r >= N) return;

  // ROI geometry (STRIDE = 8 -> * 0.125)
  const float rsw = rois[r * 5 + 1] * 0.125f;
  const float rsh = rois[r * 5 + 2] * 0.125f;
  const float rew = rois[r * 5 + 3] * 0.125f;
  const float reh = rois[r * 5 + 4] * 0.125f;
  float roi_h = reh - rsh; roi_h = (roi_h > 0.1f) ? roi_h : 0.1f;
  float roi_w = rew - rsw; roi_w = (roi_w > 0.1f) ? roi_w : 0.1f;
  const float bin_h = roi_h * (1.0f / PH_);
  const float bin_w = roi_w * (1.0f / PW_);
  const float sub_h = bin_h * (1.0f / SP_);
  const float sub_w = bin_w * (1.0f / SP_);
  const float hstart = floorf(rsh + (float)ph * bin_h);
  const float wstart = floorf(rsw + (float)pw * bin_w);

  float4 acc0 = make_float4(0.f, 0.f, 0.f, 0.f);  // channels 0..3
  float4 acc1 = make_float4(0.f, 0.f, 0.f, 0.f);  // channels 4..7
  float2 acc2 = make_float2(0.f, 0.f);            // channels 8..9
  float  count = 0.0f;

  auto gather = [&](int base, float wgt) {
    const float4 v0 = *(const float4*)&slab[base];
    const float4 v1 = *(const float4*)&slab[base + 4];
    const float2 v2 = *(const float2*)&slab[base + 8];
    acc0.x = fmaf(wgt, v0.x, acc0.x);
    acc0.y = fmaf(wgt, v0.y, acc0.y);
    acc0.z = fmaf(wgt, v0.z, acc0.z);
    acc0.w = fmaf(wgt, v0.w, acc0.w);
    acc1.x = fmaf(wgt, v1.x, acc1.x);
    acc1.y = fmaf(wgt, v1.y, acc1.y);
    acc1.z = fmaf(wgt, v1.z, acc1.z);
    acc1.w = fmaf(wgt, v1.w, acc1.w);
    acc2.x = fmaf(wgt, v2.x, acc2.x);
    acc2.y = fmaf(wgt, v2.y, acc2.y);
  };

  for (int sy = 0; sy < SP_; ++sy) {
    const float H = hstart + ((float)sy + 0.5f) * sub_h;
    const bool  keepy = (H > -1.0f) && (H < (float)FH_);
    const float y1f = floorf(H);
    const float y2f = ceilf(H);
    const float dy  = H - y1f;
    const int y1 = (int)y1f, y2 = (int)y2f;
    const bool y1v = (y1 >= 0) && (y1 < FH_);
    const bool y2v = (y2 >= 0) && (y2 < FH_);
    const int y1c = min(max(y1, 0), FH_ - 1);
    const int y2c = min(max(y2, 0), FH_ - 1);

    for (int sx = 0; sx < SP_; ++sx) {
      const float W = wstart + ((float)sx + 0.5f) * sub_w;
      const bool keep = keepy && (W > -1.0f) && (W < (float)FW_);
      if (!keep) continue;
      count += 1.0f;

      const float x1f = floorf(W);
      const float x2f = ceilf(W);
      const float dx  = W - x1f;
      const int x1 = (int)x1f, x2 = (int)x2f;
      const bool x1v = (x1 >= 0) && (x1 < FW_);
      const bool x2v = (x2 >= 0) && (x2 < FW_);
      const int x1c = min(max(x1, 0), FW_ - 1);
      const int x2c = min(max(x2, 0), FW_ - 1);

      const float w11 = (x1v && y1v) ? (1.0f - dx) * (1.0f - dy) : 0.0f;
      const float w21 = (x1v && y2v) ? (1.0f - dx) * dy          : 0.0f;
      const float w12 = (x2v && y1v) ? dx * (1.0f - dy)          : 0.0f;
      const float w22 = (x2v && y2v) ? dx * dy                   : 0.0f;

      gather((y1c * FW_ + x1c) * CPAD, w11);
      gather((y2c * FW_ + x1c) * CPAD, w21);
      gather((y1c * FW_ + x2c) * CPAD, w12);
      gather((y2c * FW_ + x2c) * CPAD, w22);
    }
  }

  // Output layout: out[(n*C + c)*49 + p]
  float* o = out + (size_t)r * C_ * BINS + p;
  if (count > 0.0f) {
    const float inv = 1.0f / count;
    o[0 * BINS] = acc0.x * inv;
    o[1 * BINS] = acc0.y * inv;
    o[2 * BINS] = acc0.z * inv;
    o[3 * BINS] = acc0.w * inv;
    o[4 * BINS] = acc1.x * inv;
    o[5 * BINS] = acc1.y * inv;
    o[6 * BINS] = acc1.z * inv;
    o[7 * BINS] = acc1.w * inv;
    o[8 * BINS] = acc2.x * inv;
    o[9 * BINS] = acc2.y * inv;
  } else {
    #pragma unroll
    for (int c = 0; c < C_; ++c) o[c * BINS] = 0.0f;
  }
}

extern "C" void kernel_launch(void* const* d_in, const int* in_sizes, int n_in,
                              void* d_out, int out_size, void* d_ws, size_t ws_size,
                              hipStream_t stream) {
  const float* ft   = (const float*)d_in[0];   // (1, 490, 34, 34) f32
  const float* rois = (const float*)d_in[1];   // (N, 5) f32
  float* out = (float*)d_out;                  // (N, 10, 49) f32
  const int N = in_sizes[1] / 5;

  dim3 grid((N + TPB - 1) / TPB, BINS);
  psroi_align_kernel<<<grid, dim3(TPB), 0, stream>>>(ft, rois, out, N);
}